// ToDenseMink_16389595201607
// MI455X (gfx1250) — compile-verified
//
#include <hip/hip_runtime.h>

// Problem constants (fixed by the reference).
#define BB 4
#define CC 256
#define XX 256
#define YY 256
#define YT 32            // y-tile per block
#define FSTR 260         // LDS row stride in floats: 260*4 = 1040 B (16B aligned, 2-way bank conflict only)

#if __has_builtin(__builtin_amdgcn_global_load_async_to_lds_b128)
#define HAVE_ASYNC_LDS 1
#else
#define HAVE_ASYNC_LDS 0
#endif

typedef int v4i __attribute__((ext_vector_type(4)));
typedef v4i* gv4i_t;                                        // generic pointer to int4
typedef __attribute__((address_space(3))) v4i* lv4i_t;      // LDS pointer to int4

static __device__ inline void wait_async_zero() {
#if __has_builtin(__builtin_amdgcn_s_wait_asynccnt)
    __builtin_amdgcn_s_wait_asynccnt(0);
#else
    asm volatile("s_wait_asynccnt 0" ::: "memory");
#endif
}

// Kernel 1: init inverse map to -1.
__global__ __launch_bounds__(256) void mink_init_map(int* __restrict__ map) {
    int i = blockIdx.x * 256 + threadIdx.x;
    if (i < BB * XX * YY) map[i] = -1;
}

// Kernel 2: scatter row indices into the map (coords are unique -> race-free).
__global__ __launch_bounds__(256) void mink_scatter_idx(const int* __restrict__ coords,
                                                        int* __restrict__ map, int n) {
    int i = blockIdx.x * 256 + threadIdx.x;
    if (i >= n) return;
    int b = coords[3 * i + 0];
    int x = coords[3 * i + 1];
    int y = coords[3 * i + 2];
    if ((unsigned)b < BB && (unsigned)x < XX && (unsigned)y < YY)
        map[(b * XX + x) * YY + y] = i;
}

// Kernel 3: tiled gather. Block = (b, x, y0..y0+31) x all 256 channels.
// Stage 32 feature rows into LDS with async global->LDS b128 copies (coalesced,
// each feats line read once), then emit fully coalesced 128B stores per channel.
__global__ __launch_bounds__(256) void mink_gather(const float* __restrict__ feats,
                                                   const int* __restrict__ map,
                                                   float* __restrict__ out) {
    __shared__ int rowIdx[YT];
    __shared__ float tile[YT * FSTR];

    const int blk = blockIdx.x;             // 0 .. 4*256*8 - 1
    const int yb  = blk & 7;
    const int x   = (blk >> 3) & 255;
    const int b   = blk >> 11;
    const int y0  = yb * YT;
    const int t   = threadIdx.x;

    if (t < YT) rowIdx[t] = map[(b * XX + x) * YY + y0 + t];
    __syncthreads();

    // Stage: 32 rows * 64 16B-chunks = 2048 chunks, 8 per thread.
    // Warp-consecutive chunks share a row -> coalesced global reads.
    bool used_async = false;
    for (int k = 0; k < 8; ++k) {
        int chunk = t + 256 * k;
        int r  = chunk >> 6;        // row in tile
        int cq = chunk & 63;        // 16B chunk within row
        int ri = rowIdx[r];
        float4* ldst = (float4*)&tile[r * FSTR] + cq;   // 16B aligned (FSTR*4 % 16 == 0)
        if (ri >= 0) {
            const float4* g = (const float4*)(feats + (size_t)ri * CC) + cq;
#if HAVE_ASYNC_LDS
            __builtin_amdgcn_global_load_async_to_lds_b128((gv4i_t)g, (lv4i_t)ldst, 0, 0);
            used_async = true;
#else
            *ldst = *g;
#endif
        } else {
            *ldst = make_float4(0.f, 0.f, 0.f, 0.f);    // hole -> zeros
        }
    }
    if (used_async) wait_async_zero();
    __syncthreads();

    // Drain: warp w handles channels c = w + 8*j, lanes sweep 32 consecutive y.
    // Every global store is a full contiguous 128B transaction.
    const int w    = t >> 5;
    const int lane = t & 31;
    for (int j = 0; j < 32; ++j) {
        int c = w + 8 * j;
        out[(((size_t)b * CC + c) * XX + x) * YY + y0 + lane] = tile[lane * FSTR + c];
    }
}

extern "C" void kernel_launch(void* const* d_in, const int* in_sizes, int n_in,
                              void* d_out, int out_size, void* d_ws, size_t ws_size,
                              hipStream_t stream) {
    const float* feats  = (const float*)d_in[0];
    const int*   coords = (const int*)d_in[1];
    float*       out    = (float*)d_out;
    int*         map    = (int*)d_ws;            // needs BB*XX*YY*4 = 1 MB
    const int n = in_sizes[1] / 3;

    const int mapElems = BB * XX * YY;
    mink_init_map<<<(mapElems + 255) / 256, 256, 0, stream>>>(map);
    mink_scatter_idx<<<(n + 255) / 256, 256, 0, stream>>>(coords, map, n);
    mink_gather<<<BB * XX * (YY / YT), 256, 0, stream>>>(feats, map, out);
}